// RobustListLearner_24249385353754
// MI455X (gfx1250) — compile-verified
//
#include <hip/hip_runtime.h>

#define N_SAMPLES 80
#define D_FEAT    56
#define S_PAIRS   3160   // C(80,2)
#define F_PAIRS   1540   // C(56,2)
#define MARGIN    0.1f

typedef unsigned int       u32;
typedef unsigned long long u64;
typedef u32 u32x4 __attribute__((ext_vector_type(4)));
typedef int i32x8 __attribute__((ext_vector_type(8)));
typedef int i32x4 __attribute__((ext_vector_type(4)));

// Low 32 bits of a flat LDS-aperture pointer are the LDS byte offset
__device__ __forceinline__ u32 lds_byte_offset(const void* p) {
    return (u32)(size_t)p;
}

// Issue one TDM 2D tile load (global -> LDS), 4-byte elements.
// D# layout per CDNA5 ISA 8.3/8.4: group0 = {count|flags, lds_addr,
// global_addr[31:0], global_addr[56:32]|type=2}; group1 packs data_size,
// tensor dims, tile dims, dim0 stride. Groups 2/3 zero (<=2D tensor).
// This toolchain's builtin is the 6-arg clang-23 form:
// (u32x4, i32x8, i32x4, i32x4, i32x8, i32 cpol).
__device__ __forceinline__ void tdm_load_2d(u32 lds_off, const void* gptr,
                                            u32 tdim0, u32 tdim1,
                                            u32 tile0, u32 tile1,
                                            u64 stride0_elems) {
    u64 ga = (u64)(size_t)gptr;
    u32x4 g0;
    g0[0] = 1u;                                              // count=1, user D#
    g0[1] = lds_off;                                         // lds_addr (bytes)
    g0[2] = (u32)ga;                                         // global_addr[31:0]
    g0[3] = ((u32)(ga >> 32) & 0x01FFFFFFu) | 0x80000000u;   // ga[56:32] | type=2

    i32x8 g1;
    g1[0] = (int)(2u << 16);                                 // data_size=2 -> 4B
    g1[1] = (int)((tdim0 & 0xFFFFu) << 16);                  // tensor_dim0[15:0] @63:48
    g1[2] = (int)(((tdim0 >> 16) & 0xFFFFu) | ((tdim1 & 0xFFFFu) << 16));
    g1[3] = (int)(((tdim1 >> 16) & 0xFFFFu) | ((tile0 & 0xFFFFu) << 16));
    g1[4] = (int)(tile1 & 0xFFFFu);                          // tile_dim1; tile_dim2=0
    g1[5] = (int)(u32)(stride0_elems & 0xFFFFFFFFull);       // dim0 stride (elems)
    g1[6] = (int)(u32)((stride0_elems >> 32) & 0xFFFFull);
    g1[7] = 0;

    i32x4 z4 = (i32x4)0;
    i32x8 z8 = (i32x8)0;
    __builtin_amdgcn_tensor_load_to_lds(g0, g1, z4, z4, z8, 0);  // TENSORcnt++
}

// Decode linear pair index into (lo, hi) for lexicographic triu_indices(n, k=1).
__device__ __forceinline__ void decode_pair(int idx, int n, int total,
                                            int& lo, int& hi) {
    int r = (total - 1) - idx;
    int k = (int)((1.0f + sqrtf((float)(8 * r + 1))) * 0.5f);
    if (k < 1) k = 1;
    while (k * (k - 1) / 2 > r) --k;       // exact integer fixup
    while (k * (k + 1) / 2 <= r) ++k;
    lo = (n - 1) - k;
    int base = total - k * (k + 1) / 2;
    hi = lo + 1 + (idx - base);
}

__global__ __launch_bounds__(256) void robust_list_pinv_kernel(
    const int* __restrict__ labels,
    const float* __restrict__ features,
    float2* __restrict__ out) {
    __shared__ float shP[N_SAMPLES];   // features[:, p]
    __shared__ float shQ[N_SAMPLES];   // features[:, q]
    __shared__ int   shL[N_SAMPLES];   // labels

    // One block per feature pair f -> (p, q); block-uniform (SGPR) decode.
    int p, q;
    decode_pair((int)blockIdx.x, D_FEAT, F_PAIRS, p, q);

    // Wave 0 issues three tiny TDM DMAs (TDM ignores EXEC -> one wave only).
    if (threadIdx.x < 32u) {
        tdm_load_2d(lds_byte_offset(shP), features + p,
                    D_FEAT, N_SAMPLES, /*tile*/ 1, N_SAMPLES, D_FEAT);
        tdm_load_2d(lds_byte_offset(shQ), features + q,
                    D_FEAT, N_SAMPLES, /*tile*/ 1, N_SAMPLES, D_FEAT);
        tdm_load_2d(lds_byte_offset(shL), labels,
                    N_SAMPLES, 1, /*tile*/ N_SAMPLES, 0, N_SAMPLES);
        __builtin_amdgcn_s_wait_tensorcnt(0);
    }
    __syncthreads();

    const u32 mbase = (u32)blockIdx.x * (u32)S_PAIRS;

    for (int s = (int)threadIdx.x; s < S_PAIRS; s += (int)blockDim.x) {
        int i, j;
        decode_pair(s, N_SAMPLES, S_PAIRS, i, j);

        const float yi = (float)(2 * shL[i] - 1);
        const float yj = (float)(2 * shL[j] - 1);

        // A = [[lf[i,p], lf[i,q]], [lf[j,p], lf[j,q]]],  lf = y * features
        const float a00 = yi * shP[i];
        const float a01 = yi * shQ[i];
        const float a10 = yj * shP[j];
        const float a11 = yj * shQ[j];
        const float r0 = yi - MARGIN;
        const float r1 = yj - MARGIN;

        const float det   = fmaf(a00, a11, -(a01 * a10));
        const float frob2 = fmaf(a00, a00, fmaf(a01, a01,
                            fmaf(a10, a10, a11 * a11)));

        // Full-rank: w = inv(A) rhs (Cramer). Near-singular (SVD cutoff
        // ~ 2*eps_f32 * sigma_max): rank-1 pinv -> w = A^T rhs / ||A||_F^2.
        const bool ok = fabsf(det) > 2.4e-7f * frob2;
        const float num0 = ok ? fmaf(a11, r0, -(a01 * r1))
                              : fmaf(a00, r0,   a10 * r1);
        const float num1 = ok ? fmaf(a00, r1, -(a10 * r0))
                              : fmaf(a01, r0,   a11 * r1);
        const float den  = ok ? det : frob2;
        const float inv  = (den != 0.0f) ? (1.0f / den) : 0.0f;

        // m = f*S + s; consecutive lanes -> consecutive float2 (b64, coalesced)
        out[mbase + (u32)s] = make_float2(num0 * inv, num1 * inv);
    }
}

extern "C" void kernel_launch(void* const* d_in, const int* in_sizes, int n_in,
                              void* d_out, int out_size, void* d_ws, size_t ws_size,
                              hipStream_t stream) {
    const int*   labels   = (const int*)d_in[0];
    const float* features = (const float*)d_in[1];
    float2*      out      = (float2*)d_out;

    dim3 grid(F_PAIRS);   // one block per feature pair
    dim3 block(256);      // 8 wave32 waves
    hipLaunchKernelGGL(robust_list_pinv_kernel, grid, block, 0, stream,
                       labels, features, out);
}